// NucleusFilter_11957188952196
// MI455X (gfx1250) — compile-verified
//
#include <hip/hip_runtime.h>
#include <stdint.h>

// Nucleus (top-p) selection for B rows of N=50257 softmax weights.
//   out = [ sorted_idx (B*N int32) | norm_w (B*N f32) | k (B int32) ]
//
// Algorithm: per-row stable LSD radix sort on key = ~float_bits(w)
// (monotone for w in [0,2), only bits [29:0] vary) -> 6 passes x 5 bits,
// one workgroup per row, then a tiled scan to find k and normalize.

#define VOCAB_N 50257
#define BLK     256        // 8 wave32s per workgroup
#define RBITS   5
#define RADIX   32
#define RSTRIDE 33         // pad to avoid LDS bank conflicts
#define ITEMS   4
#define TILE    (BLK * ITEMS)
#define NPASS   6          // 6*5 = 30 bits (bits 31:30 constant for w < 2.0)
#define P_THRESH 0.4f

#if defined(__has_builtin)
#if __has_builtin(__builtin_amdgcn_global_load_async_to_lds_b32) && \
    __has_builtin(__builtin_amdgcn_s_wait_asynccnt)
#define USE_ASYNC_LDS 1
#endif
#endif

#if defined(USE_ASYNC_LDS)
typedef __attribute__((address_space(1))) int async_gptr_t;   // global src
typedef __attribute__((address_space(3))) int async_lptr_t;   // LDS dst
#endif

// ---------------------------------------------------------------- prep ----
__global__ void nucleus_prep(const float* __restrict__ w,
                             uint32_t* __restrict__ keyA,
                             uint32_t* __restrict__ idxA, int N) {
  const size_t base = (size_t)blockIdx.x * (size_t)N;
  for (int i = threadIdx.x; i < N; i += BLK) {
    __builtin_prefetch(&w[base + i + 4 * BLK], 0, 1);
    uint32_t b = __float_as_uint(w[base + i]);
    keyA[base + i] = ~b;   // ascending(~bits) == stable descending(w), w >= 0
    idxA[base + i] = (uint32_t)i;
  }
}

// ---------------------------------------------------------- radix pass ----
__global__ void nucleus_radix_pass(const uint32_t* __restrict__ kin,
                                   const uint32_t* __restrict__ vin,
                                   uint32_t* __restrict__ kout,
                                   uint32_t* __restrict__ vout,
                                   int N, int shift) {
  __shared__ uint32_t s_whist[8][RADIX];       // per-wave32 histograms
  __shared__ uint32_t s_offs[RADIX];           // running global digit offsets
  __shared__ uint32_t s_tot[RADIX];            // per-tile digit totals
  __shared__ uint32_t s_cnt[BLK * RSTRIDE];    // per-thread digit counts
  const int tid  = threadIdx.x;
  const int wv   = tid >> 5;                   // wave32 index (warpSize==32)
  const size_t base = (size_t)blockIdx.x * (size_t)N;

  // Phase A: row-wide digit histogram
  if (tid < RADIX) {
#pragma unroll
    for (int q = 0; q < 8; ++q) s_whist[q][tid] = 0u;
  }
  __syncthreads();
  for (int i = tid; i < N; i += BLK) {
    __builtin_prefetch(&kin[base + i + 4 * BLK], 0, 1);
    uint32_t k = kin[base + i];
    atomicAdd(&s_whist[wv][(k >> shift) & (RADIX - 1)], 1u);
  }
  __syncthreads();
  if (tid == 0) {
    uint32_t run = 0;
    for (int d = 0; d < RADIX; ++d) {
      uint32_t c = 0;
      for (int q = 0; q < 8; ++q) c += s_whist[q][d];
      s_offs[d] = run;
      run += c;
    }
  }
  __syncthreads();

  // Phase B: stable tiled scatter
  const int ntiles = (N + TILE - 1) / TILE;
  for (int t = 0; t < ntiles; ++t) {
    const int tstart = t * TILE;
#pragma unroll
    for (int d = 0; d < RADIX; ++d) s_cnt[tid * RSTRIDE + d] = 0u;

    uint32_t k[ITEMS], v[ITEMS];
    int dg[ITEMS], pr[ITEMS];
    bool ok[ITEMS];
    const int ebase = tstart + tid * ITEMS;   // blocked arrangement => stable
#pragma unroll
    for (int j = 0; j < ITEMS; ++j) {
      int i = ebase + j;
      ok[j] = (i < N);
      k[j]  = ok[j] ? kin[base + i] : 0u;
      v[j]  = ok[j] ? vin[base + i] : 0u;
      dg[j] = (int)((k[j] >> shift) & (RADIX - 1));
    }
    __builtin_prefetch(&kin[base + ebase + TILE], 0, 1);
#pragma unroll
    for (int j = 0; j < ITEMS; ++j) {
      int p = 0;
#pragma unroll
      for (int jj = 0; jj < ITEMS; ++jj)
        if (jj < j) p += (ok[jj] && dg[jj] == dg[j]) ? 1 : 0;
      pr[j] = p;
      if (ok[j]) s_cnt[tid * RSTRIDE + dg[j]] += 1u;
    }
    __syncthreads();

    // column-wise exclusive scan over threads (one thread per digit)
    if (tid < RADIX) {
      uint32_t run = 0;
      for (int tt = 0; tt < BLK; ++tt) {
        uint32_t c = s_cnt[tt * RSTRIDE + tid];
        s_cnt[tt * RSTRIDE + tid] = run;
        run += c;
      }
      s_tot[tid] = run;
    }
    __syncthreads();

#pragma unroll
    for (int j = 0; j < ITEMS; ++j) {
      if (ok[j]) {
        uint32_t r = s_offs[dg[j]] + s_cnt[tid * RSTRIDE + dg[j]] +
                     (uint32_t)pr[j];
        kout[base + r] = k[j];
        vout[base + r] = v[j];
      }
    }
    __syncthreads();
    if (tid < RADIX) s_offs[tid] += s_tot[tid];
    __syncthreads();
  }
}

// ------------------------------------------------------------ finalize ----
__global__ void nucleus_finalize(const uint32_t* __restrict__ kA,
                                 const uint32_t* __restrict__ vA, int N,
                                 int* __restrict__ out_idx,
                                 float* __restrict__ out_w,
                                 int* __restrict__ out_k) {
  __shared__ float    s_scan[BLK];
  __shared__ uint32_t s_key[BLK];
  __shared__ int      s_minpos;
  __shared__ float    s_carry;
  __shared__ int      s_k;
  __shared__ float    s_sum;
  const int tid  = threadIdx.x;
  const size_t base = (size_t)blockIdx.x * (size_t)N;

  if (tid == 0) { s_minpos = 0x7FFFFFFF; s_carry = 0.0f; s_k = -1; s_sum = 0.0f; }
  __syncthreads();

  const int ntiles = (N + BLK - 1) / BLK;
  for (int t = 0; t < ntiles; ++t) {
    const int  i    = t * BLK + tid;
    const bool full = ((t + 1) * BLK <= N);
#if defined(USE_ASYNC_LDS)
    if (full) {
      // CDNA5 async global->LDS copy (ASYNCcnt path)
      __builtin_amdgcn_global_load_async_to_lds_b32(
          (async_gptr_t*)(kA + base + i),
          (async_lptr_t*)&s_key[tid], 0, 0);
      __builtin_amdgcn_s_wait_asynccnt(0);
    } else {
      s_key[tid] = (i < N) ? kA[base + i] : 0xFFFFFFFFu;
    }
#else
    s_key[tid] = (i < N) ? kA[base + i] : 0xFFFFFFFFu;
#endif
    __builtin_prefetch(kA + base + i + BLK, 0, 1);
    __syncthreads();

    const float x = (i < N) ? __uint_as_float(~s_key[tid]) : 0.0f;
    s_scan[tid] = x;
    __syncthreads();
#pragma unroll
    for (int off = 1; off < BLK; off <<= 1) {
      float y = (tid >= off) ? s_scan[tid - off] : 0.0f;
      __syncthreads();
      s_scan[tid] += y;
      __syncthreads();
    }
    const float cums = s_carry + s_scan[tid];
    if (i < N && cums > P_THRESH) atomicMin(&s_minpos, i);
    __syncthreads();
    if (s_minpos != 0x7FFFFFFF) {          // uniform decision
      if (i == s_minpos) { s_sum = cums; s_k = s_minpos + 1; }
      __syncthreads();
      break;
    }
    if (tid == 0) s_carry += s_scan[BLK - 1];
    __syncthreads();
  }
  if (tid == 0 && s_k < 0) { s_k = N; s_sum = s_carry; }  // never exceeded
  __syncthreads();

  const int   kk  = s_k;
  const float sum = s_sum;
  for (int i = tid; i < N; i += BLK) {
    out_idx[base + i] = (int)vA[base + i];
    const float wv = __uint_as_float(~kA[base + i]);
    out_w[base + i] = (i < kk) ? (wv / sum) : 0.0f;
  }
  if (tid == 0) out_k[blockIdx.x] = kk;
}

// -------------------------------------------------------------- launch ----
extern "C" void kernel_launch(void* const* d_in, const int* in_sizes, int n_in,
                              void* d_out, int out_size, void* d_ws,
                              size_t ws_size, hipStream_t stream) {
  (void)n_in; (void)out_size; (void)ws_size;
  const float* w  = (const float*)d_in[0];
  const int    N  = VOCAB_N;
  const int    BN = in_sizes[0];
  const int    B  = BN / N;

  // workspace: ping-pong key/index buffers (4 * BN * 4 bytes)
  uint32_t* keyA = (uint32_t*)d_ws;
  uint32_t* idxA = keyA + (size_t)BN;
  uint32_t* keyB = idxA + (size_t)BN;
  uint32_t* idxB = keyB + (size_t)BN;

  int*   out_idx = (int*)d_out;
  float* out_w   = (float*)d_out + (size_t)BN;
  int*   out_k   = (int*)((float*)d_out + 2 * (size_t)BN);

  nucleus_prep<<<B, BLK, 0, stream>>>(w, keyA, idxA, N);

  const uint32_t* kin = keyA;
  const uint32_t* vin = idxA;
  uint32_t*       kout = keyB;
  uint32_t*       vout = idxB;
  for (int p = 0; p < NPASS; ++p) {
    nucleus_radix_pass<<<B, BLK, 0, stream>>>(kin, vin, kout, vout, N,
                                              p * RBITS);
    const uint32_t* tk = kin; kin = kout; kout = (uint32_t*)tk;
    const uint32_t* tv = vin; vin = vout; vout = (uint32_t*)tv;
  }
  // NPASS is even -> sorted result back in keyA/idxA (== kin/vin here)
  nucleus_finalize<<<B, BLK, 0, stream>>>(kin, vin, N, out_idx, out_w, out_k);
}